// GraphConvolution_46411416600780
// MI455X (gfx1250) — compile-verified
//
#include <hip/hip_runtime.h>

typedef __attribute__((ext_vector_type(2))) float v2f;
typedef __attribute__((ext_vector_type(4))) float v4f;
typedef __attribute__((ext_vector_type(8))) float v8f;

#define FIN  256
#define DOUT 64

// ---------------------------------------------------------------------------
// Kernel 1: zero the accumulator (d_out) so edge atomics can accumulate into it
// ---------------------------------------------------------------------------
__global__ void gcn_zero(float* __restrict__ out, int n4) {
    int t = blockIdx.x * blockDim.x + threadIdx.x;
    if (t < n4) {
        v4f z = {0.f, 0.f, 0.f, 0.f};
        ((v4f*)out)[t] = z;
    }
}

// ---------------------------------------------------------------------------
// Kernel 2: XW = X @ W via V_WMMA_F32_16X16X4_F32 (full-precision f32 WMMA)
//
// Block = 256 threads = 8 waves. Block tile = 32 rows x 64 cols.
//   wave (mtile, ntile): mtile in {0,1} picks a 16-row slab, ntile in {0..3}
//   a 16-col slab. K = 256 swept in 64 steps of 4, unrolled 8x so each loop
//   iteration issues an 8-deep global_load_b64 clause + 8 DS loads + 8 WMMAs
//   with staggered s_wait_loadcnt_dscnt.
//
// f32 16x4 A layout (ISA 7.12.2): lanes 0-15 hold M=lane, VGPR0/1 = K0/K1;
// lanes 16-31 hold M=lane-16, VGPR0/1 = K2/K3  -> per lane a float2 at
// x[row, k + 2*(lane>=16)]. B (4x16) mirrors this per-K-row across lanes.
// C/D: VGPR r -> (M = r + 8*(lane>=16), N = lane&15).
// ---------------------------------------------------------------------------
__global__ __launch_bounds__(256) void gcn_gemm_wmma(
    const float* __restrict__ x, const float* __restrict__ w,
    float* __restrict__ xw, int nNodes)
{
    __shared__ float wlds[FIN * DOUT];   // 64 KB: whole weight matrix

    const int tid = threadIdx.x;
    {   // stage W into LDS, float4-wide, fully coalesced
        const v4f* src = (const v4f*)w;
        v4f* dst = (v4f*)wlds;
        #pragma unroll
        for (int i = 0; i < (FIN * DOUT / 4) / 256; ++i)
            dst[i * 256 + tid] = src[i * 256 + tid];
    }
    __syncthreads();

    const int wave  = tid >> 5;
    const int lane  = tid & 31;
    const int h     = lane >> 4;    // lane half: selects K pair (A) / M+8 (C/D)
    const int l16   = lane & 15;
    const int mtile = wave >> 2;    // 0..1
    const int ntile = wave & 3;     // 0..3

    const int rowBase = blockIdx.x * 32 + mtile * 16;
    int row  = rowBase + l16;
    int rowc = row < nNodes ? row : (nNodes - 1);   // clamp tail-tile loads
    const int ncol = ntile * 16 + l16;

    const float* arow = x + (size_t)rowc * FIN + 2 * h;  // 8B aligned

    v8f acc = {};
    #pragma unroll 8
    for (int k = 0; k < FIN; k += 4) {
        v2f a = *(const v2f*)(arow + k);                 // global_load_b64
        v2f b;
        b.x = wlds[(k + 2 * h)     * DOUT + ncol];       // ds_load (2addr fused)
        b.y = wlds[(k + 2 * h + 1) * DOUT + ncol];
        acc = __builtin_amdgcn_wmma_f32_16x16x4_f32(
                  /*neg_a=*/false, a, /*neg_b=*/false, b,
                  /*c_mod=*/(short)0, acc,
                  /*reuse_a=*/false, /*reuse_b=*/false);
    }

    // Epilogue: wave-uniform fast path -> 8 stores off one base address with
    // immediate offsets; per-row guards only in the final partial block.
    float* dst = xw + (size_t)(rowBase + 8 * h) * DOUT + ncol;
    if (rowBase + 16 <= nNodes) {
        #pragma unroll
        for (int r = 0; r < 8; ++r)
            dst[r * DOUT] = acc[r];                      // global_store_b32 +imm
    } else {
        #pragma unroll
        for (int r = 0; r < 8; ++r) {
            if (rowBase + 8 * h + r < nNodes)
                dst[r * DOUT] = acc[r];
        }
    }
}

// ---------------------------------------------------------------------------
// Kernel 3: edge scatter. 16 lanes per edge, each owns a float4 of the 64-dim
// row. XW (12.8 MB) is L2-resident on MI455X (192 MB L2), so the random
// gather hits L2; accumulation via hardware global_atomic_add_f32 (no return
// -> STOREcnt, fire-and-forget).
// ---------------------------------------------------------------------------
__global__ void gcn_spmm_scatter(
    const float* __restrict__ xw, const float* __restrict__ vals,
    const int* __restrict__ rows, const int* __restrict__ cols,
    float* __restrict__ agg, int nEdges)
{
    long t = (long)blockIdx.x * blockDim.x + threadIdx.x;
    int e = (int)(t >> 4);
    int q = (int)(t & 15);
    if (e >= nEdges) return;

    float v = vals[e];
    int   c = cols[e];
    int   r = rows[e];

    v4f g = ((const v4f*)xw)[(size_t)c * 16 + q];        // L2-served gather
    float* dst = agg + (size_t)r * DOUT + q * 4;
    unsafeAtomicAdd(dst + 0, g.x * v);                   // global_atomic_add_f32
    unsafeAtomicAdd(dst + 1, g.y * v);
    unsafeAtomicAdd(dst + 2, g.z * v);
    unsafeAtomicAdd(dst + 3, g.w * v);
}

// ---------------------------------------------------------------------------
// Kernel 4: in-place ReLU on d_out
// ---------------------------------------------------------------------------
__global__ void gcn_relu(float* __restrict__ out, int n4) {
    int t = blockIdx.x * blockDim.x + threadIdx.x;
    if (t < n4) {
        v4f g = ((v4f*)out)[t];
        g.x = g.x > 0.f ? g.x : 0.f;
        g.y = g.y > 0.f ? g.y : 0.f;
        g.z = g.z > 0.f ? g.z : 0.f;
        g.w = g.w > 0.f ? g.w : 0.f;
        ((v4f*)out)[t] = g;
    }
}

// ---------------------------------------------------------------------------
extern "C" void kernel_launch(void* const* d_in, const int* in_sizes, int n_in,
                              void* d_out, int out_size, void* d_ws, size_t ws_size,
                              hipStream_t stream) {
    const float* x    = (const float*)d_in[0];   // [N, 256]
    const float* w    = (const float*)d_in[1];   // [256, 64]
    const float* vals = (const float*)d_in[2];   // [E]
    const int*   rows = (const int*)d_in[3];     // [E]
    const int*   cols = (const int*)d_in[4];     // [E]
    float*       out  = (float*)d_out;           // [N, 64]
    float*       xw   = (float*)d_ws;            // [N, 64] scratch (12.8 MB)

    const int nNodes = in_sizes[0] / FIN;
    const int nEdges = in_sizes[2];

    // 1) out = 0
    {
        int n4 = out_size / 4;
        int blocks = (n4 + 255) / 256;
        gcn_zero<<<blocks, 256, 0, stream>>>(out, n4);
    }
    // 2) xw = x @ w  (WMMA f32)
    {
        int blocks = (nNodes + 31) / 32;
        gcn_gemm_wmma<<<blocks, 256, 0, stream>>>(x, w, xw, nNodes);
    }
    // 3) out[r] += vals[e] * xw[c]   (atomics)
    {
        long threads = (long)nEdges * 16;
        int blocks = (int)((threads + 255) / 256);
        gcn_spmm_scatter<<<blocks, 256, 0, stream>>>(xw, vals, rows, cols, out, nEdges);
    }
    // 4) out = relu(out), in place
    {
        int n4 = out_size / 4;
        int blocks = (n4 + 255) / 256;
        gcn_relu<<<blocks, 256, 0, stream>>>(out, n4);
    }
}